// Encoder_DGI_5385888989905
// MI455X (gfx1250) — compile-verified
//
#include <hip/hip_runtime.h>
#include <hip/hip_bf16.h>

// ---------------------------------------------------------------------------
// GCN encoder (DGI): spectral-norm linear + GCNConv aggregation + PReLU.
// Memory-bound on MI455X; GEMM done with V_WMMA_F32_16X16X4_F32 (exact f32).
// ---------------------------------------------------------------------------

typedef __attribute__((ext_vector_type(2))) float v2f;
typedef __attribute__((ext_vector_type(8))) float v8f;

#define F_IN 128
#define F_OUT 32

// --- K0: sigma = ||W v||, v = l2norm(W^T u); write 1/sigma ------------------
__global__ void __launch_bounds__(128)
dgi_sigma_kernel(const float* __restrict__ W, const float* __restrict__ u,
                 float* __restrict__ inv_sigma) {
    __shared__ float v[F_IN];
    __shared__ float red[F_IN];
    const int t = threadIdx.x;  // 0..127
    // v_raw[t] = sum_i W[i][t] * u[i]
    float acc = 0.0f;
    #pragma unroll 8
    for (int i = 0; i < F_OUT; ++i) acc += W[i * F_IN + t] * u[i];
    v[t] = acc;
    red[t] = acc * acc;
    __syncthreads();
    for (int s = 64; s > 0; s >>= 1) {
        if (t < s) red[t] += red[t + s];
        __syncthreads();
    }
    const float vn = fmaxf(sqrtf(red[0]), 1e-12f);
    const float vt = v[t] / vn;
    __syncthreads();
    v[t] = vt;
    __syncthreads();
    // wv[i] = sum_j W[i][j] * v[j], for i < 32; sigma = ||wv||
    float wv = 0.0f;
    if (t < F_OUT) {
        #pragma unroll 8
        for (int j = 0; j < F_IN; ++j) wv += W[t * F_IN + j] * v[j];
    }
    red[t] = (t < F_OUT) ? wv * wv : 0.0f;
    __syncthreads();
    for (int s = 64; s > 0; s >>= 1) {
        if (t < s) red[t] += red[t + s];
        __syncthreads();
    }
    if (t == 0) inv_sigma[0] = 1.0f / fmaxf(sqrtf(red[0]), 1e-12f);
}

// --- K1: deg[i] = 1 (self loop) ---------------------------------------------
__global__ void dgi_deg_init_kernel(float* __restrict__ deg, int n) {
    int i = blockIdx.x * blockDim.x + threadIdx.x;
    if (i < n) deg[i] = 1.0f;
}

// --- K2: deg[col[e]] += 1 ---------------------------------------------------
__global__ void dgi_deg_kernel(const int* __restrict__ col,
                               float* __restrict__ deg, int nedges) {
    int e = blockIdx.x * blockDim.x + threadIdx.x;
    if (e < nedges) atomicAdd(&deg[col[e]], 1.0f);
}

// --- K3: dis[i] = rsqrt(deg[i]) in place ------------------------------------
__global__ void dgi_dis_kernel(float* __restrict__ deg, int n) {
    int i = blockIdx.x * blockDim.x + threadIdx.x;
    if (i < n) {
        float d = deg[i];
        deg[i] = (d > 0.0f) ? rsqrtf(d) : 0.0f;
    }
}

// --- K4: xw = x @ (W/sigma)^T via V_WMMA_F32_16X16X4_F32 --------------------
// One wave per 16-row node tile; two v8f accumulators cover 32 out features.
// A (16x4 f32) lane layout: lanes 0-15 M=lane {K=0,1}, lanes 16-31 {K=2,3}.
// B (4x16 f32) symmetric: lanes 0-15 N=lane {K=0,1}, lanes 16-31 {K=2,3}.
__global__ void __launch_bounds__(256)
dgi_gemm_kernel(const float* __restrict__ x, const float* __restrict__ W,
                const float* __restrict__ inv_sigma, float* __restrict__ xw,
                int nnodes) {
    __shared__ float Ws[F_OUT * F_IN];  // 16 KB: scaled weight, reused by 8 waves
    const float is = inv_sigma[0];
    for (int i = threadIdx.x; i < F_OUT * F_IN; i += 256) Ws[i] = W[i] * is;
    __syncthreads();

    const int wave = threadIdx.x >> 5;
    const int lane = threadIdx.x & 31;
    const int rowBase = (blockIdx.x * 8 + wave) * 16;
    if (rowBase >= nnodes) return;  // wave-uniform: EXEC stays all-1 for WMMA

    const int lh = lane >> 4;   // K sub-pair select (0: K0/K1, 1: K2/K3)
    const int lm = lane & 15;   // M (for A) / N (for B)

    const float* ap  = x  + (size_t)(rowBase + lm) * F_IN + lh * 2;
    const float* b0p = Ws + lm * F_IN + lh * 2;
    const float* b1p = Ws + (lm + 16) * F_IN + lh * 2;

    v8f c0 = {};
    v8f c1 = {};
    #pragma unroll 4
    for (int k = 0; k < F_IN; k += 4) {
        v2f a, b0, b1;
        a.x  = ap[k];  a.y  = ap[k + 1];
        b0.x = b0p[k]; b0.y = b0p[k + 1];
        b1.x = b1p[k]; b1.y = b1p[k + 1];
        c0 = __builtin_amdgcn_wmma_f32_16x16x4_f32(false, a, false, b0,
                                                   (short)0, c0, false, false);
        c1 = __builtin_amdgcn_wmma_f32_16x16x4_f32(false, a, false, b1,
                                                   (short)0, c1, false, false);
    }

    // C/D layout: VGPR r -> (M = r + lh*8, N = lm) ; second half N = lm+16.
    float* orow = xw + (size_t)rowBase * F_OUT;
    #pragma unroll
    for (int r = 0; r < 8; ++r) {
        const int row = r + lh * 8;
        orow[row * F_OUT + lm]      = c0[r];
        orow[row * F_OUT + lm + 16] = c1[r];
    }
}

// --- K5: out = xw * dis^2 (self-loop contribution, also initializes d_out) --
__global__ void dgi_selfloop_kernel(const float* __restrict__ xw,
                                    const float* __restrict__ dis,
                                    float* __restrict__ out, int total) {
    int t = blockIdx.x * blockDim.x + threadIdx.x;
    if (t < total) {
        float d = dis[t >> 5];
        out[t] = xw[t] * d * d;
    }
}

// --- K6: edge scatter: out[c] += xw[r] * dis[r]*dis[c] ----------------------
__global__ void dgi_scatter_kernel(const int* __restrict__ ei,
                                   const float* __restrict__ xw,
                                   const float* __restrict__ dis,
                                   float* __restrict__ out, int nedges) {
    int t = blockIdx.x * blockDim.x + threadIdx.x;
    if (t >= nedges * 8) return;
    const int e = t >> 3;       // edge id
    const int g = t & 7;        // 4-feature group (32/4)
    __builtin_prefetch(ei + e + 8192, 0, 0);  // global_prefetch_b8 of edge data
    const int r = ei[e];
    const int c = ei[nedges + e];
    const float nrm = dis[r] * dis[c];
    const float4 m = ((const float4*)(xw + (size_t)r * F_OUT))[g];
    float* dst = out + (size_t)c * F_OUT + g * 4;
    atomicAdd(dst + 0, m.x * nrm);
    atomicAdd(dst + 1, m.y * nrm);
    atomicAdd(dst + 2, m.z * nrm);
    atomicAdd(dst + 3, m.w * nrm);
}

// --- K7: out = PReLU(out + bias) --------------------------------------------
__global__ void dgi_finalize_kernel(float* __restrict__ out,
                                    const float* __restrict__ bias,
                                    const float* __restrict__ prelu_a,
                                    int total) {
    int t = blockIdx.x * blockDim.x + threadIdx.x;
    if (t < total) {
        const float v = out[t] + bias[t & 31];
        const float a = prelu_a[0];
        out[t] = (v >= 0.0f) ? v : a * v;
    }
}

extern "C" void kernel_launch(void* const* d_in, const int* in_sizes, int n_in,
                              void* d_out, int out_size, void* d_ws, size_t ws_size,
                              hipStream_t stream) {
    const float* x       = (const float*)d_in[0];   // [N, 128]
    const int*   ei      = (const int*)d_in[1];     // [2, E] flattened
    const float* W       = (const float*)d_in[2];   // [32, 128]
    const float* bias    = (const float*)d_in[3];   // [32]
    const float* prelu_a = (const float*)d_in[4];   // [1]
    const float* u_sn    = (const float*)d_in[5];   // [32]
    float* out = (float*)d_out;

    const int N = in_sizes[0] / F_IN;
    const int E = in_sizes[1] / 2;

    // workspace layout (floats): [0..15] inv_sigma, [16..16+N) deg/dis,
    // [16+N .. 16+N+32N) xw
    float* wsf       = (float*)d_ws;
    float* inv_sigma = wsf;
    float* deg       = wsf + 16;
    float* xw        = wsf + 16 + N;

    const int B = 256;

    dgi_sigma_kernel<<<1, 128, 0, stream>>>(W, u_sn, inv_sigma);
    dgi_deg_init_kernel<<<(N + B - 1) / B, B, 0, stream>>>(deg, N);
    dgi_deg_kernel<<<(E + B - 1) / B, B, 0, stream>>>(ei + E, deg, E);
    dgi_dis_kernel<<<(N + B - 1) / B, B, 0, stream>>>(deg, N);

    const int tiles = (N + 15) / 16;           // 6250 full tiles (N % 16 == 0)
    dgi_gemm_kernel<<<(tiles + 7) / 8, B, 0, stream>>>(x, W, inv_sigma, xw, N);

    const int total = N * F_OUT;
    dgi_selfloop_kernel<<<(total + B - 1) / B, B, 0, stream>>>(xw, deg, out, total);

    const long long sthreads = (long long)E * 8;
    dgi_scatter_kernel<<<(int)((sthreads + B - 1) / B), B, 0, stream>>>(ei, xw, deg, out, E);

    dgi_finalize_kernel<<<(total + B - 1) / B, B, 0, stream>>>(out, bias, prelu_a, total);
}